// GlobalClusteringModule_395136991789
// MI455X (gfx1250) — compile-verified
//
#include <hip/hip_runtime.h>
#include <float.h>
#include <math.h>

// ---------------- problem constants ----------------
#define B_   128
#define N_   300
#define DV_  1024
#define DA_  128
#define K_   64          // CLUSTER_COUNT
#define EPS_ 1e-5f
#define GP_  304         // padded Gram row stride
#define NM_  10          // 32-wide macro tiles (covers 320 >= 300)
#define NPAIR_ 55        // NM_*(NM_+1)/2 upper-triangle macro pairs

typedef float v2f __attribute__((ext_vector_type(2)));
typedef float v8f __attribute__((ext_vector_type(8)));

// ---------------- kernel 1: inverse norms ----------------
// one wave per row; 8 rows per 256-thread block
__global__ void __launch_bounds__(256) knorm(const float* __restrict__ video,
                                             float* __restrict__ inv) {
  int row  = blockIdx.x * 8 + (threadIdx.x >> 5);
  int lane = threadIdx.x & 31;
  if (row >= B_ * N_) return;
  const float* p = video + (size_t)row * DV_;
  float s = 0.f;
  for (int e = lane; e < DV_; e += 32) { float x = p[e]; s += x * x; }
  for (int m = 16; m; m >>= 1) s += __shfl_xor(s, m, 32);
  if (lane == 0) inv[row] = 1.0f / (sqrtf(s) + EPS_);
}

// ---------------- Gram tile store helper ----------------
__device__ __forceinline__ void store_tile(float* __restrict__ Gb,
                                           const float* __restrict__ invb,
                                           int ibase, int j, float invj, int h,
                                           v8f acc) {
  if (j >= N_) return;
#pragma unroll
  for (int v = 0; v < 8; ++v) {
    int i = ibase + v + 8 * h;               // C/D layout: (m = v+8h, n = l)
    if (i < N_) Gb[(size_t)i * GP_ + j] = fabsf(acc[v]) * invb[i] * invj;
  }
}

// ---------------- kernel 2: batched symmetric Gram via f32 WMMA ----------------
// one wave per 32x32 macro-tile (2x2 WMMA tiles), upper triangle only.
// grid (ceil(55/8), B), 256 threads (8 waves).
// Stores G[i][j] for macro-row<=macro-col; consumer reads (min,max).
__global__ void __launch_bounds__(256) kgram(const float* __restrict__ video,
                                             const float* __restrict__ inv,
                                             float* __restrict__ G) {
  int wave = threadIdx.x >> 5;
  int lane = threadIdx.x & 31;
  int pair = blockIdx.x * 8 + wave;
  if (pair >= NPAIR_) return;                // wave-uniform exit, EXEC stays full
  int mi = 0, t = pair;                      // map linear id -> (mi<=mj)
  while (t >= NM_ - mi) { t -= NM_ - mi; ++mi; }
  int mj = mi + t;
  int b = blockIdx.y;
  int l = lane & 15, h = lane >> 4;

  int rA0 = mi * 32 + l;      if (rA0 >= N_) rA0 = N_ - 1;   // clamp pad rows
  int rA1 = mi * 32 + 16 + l; if (rA1 >= N_) rA1 = N_ - 1;
  int rB0 = mj * 32 + l;      if (rB0 >= N_) rB0 = N_ - 1;
  int rB1 = mj * 32 + 16 + l; if (rB1 >= N_) rB1 = N_ - 1;

  const float* base = video + (size_t)b * N_ * DV_ + 2 * h;
  const float* pA0 = base + (size_t)rA0 * DV_;
  const float* pA1 = base + (size_t)rA1 * DV_;
  const float* pB0 = base + (size_t)rB0 * DV_;
  const float* pB1 = base + (size_t)rB1 * DV_;

  v8f acc00 = {}, acc01 = {}, acc10 = {}, acc11 = {};
#pragma unroll 4
  for (int k = 0; k < DV_; k += 4) {
    v2f a0 = *(const v2f*)(pA0 + k);
    v2f a1 = *(const v2f*)(pA1 + k);
    v2f b0 = *(const v2f*)(pB0 + k);
    v2f b1 = *(const v2f*)(pB1 + k);
    acc00 = __builtin_amdgcn_wmma_f32_16x16x4_f32(false, a0, false, b0, (short)0, acc00, false, false);
    acc01 = __builtin_amdgcn_wmma_f32_16x16x4_f32(false, a0, false, b1, (short)0, acc01, false, false);
    acc10 = __builtin_amdgcn_wmma_f32_16x16x4_f32(false, a1, false, b0, (short)0, acc10, false, false);
    acc11 = __builtin_amdgcn_wmma_f32_16x16x4_f32(false, a1, false, b1, (short)0, acc11, false, false);
  }

  float* Gb = G + (size_t)b * GP_ * GP_;
  const float* invb = inv + b * N_;
  int j0 = mj * 32 + l;
  int j1 = mj * 32 + 16 + l;
  float invj0 = (j0 < N_) ? invb[j0] : 0.f;
  float invj1 = (j1 < N_) ? invb[j1] : 0.f;
  store_tile(Gb, invb, mi * 32,      j0, invj0, h, acc00);
  store_tile(Gb, invb, mi * 32,      j1, invj1, h, acc01);
  store_tile(Gb, invb, mi * 32 + 16, j0, invj0, h, acc10);
  store_tile(Gb, invb, mi * 32 + 16, j1, invj1, h, acc11);
}

// ---------------- kernel 3: greedy farthest-point selection ----------------
// one 320-thread block per batch; thread t owns point t (t < 300).
// G stored only for macro-row<=macro-col: read element (min(r,c), max(r,c)).
__global__ void __launch_bounds__(320) ksel(const float* __restrict__ G,
                                            int* __restrict__ idx) {
  __shared__ float s_val[512];
  __shared__ int   s_idx[512];
  __shared__ int   s_sel[K_];
  int b = blockIdx.x, tid = threadIdx.x;
  const float* Gb = G + (size_t)b * GP_ * GP_;
  bool valid = tid < N_;

  for (int i = tid + 320; i < 512; i += 320) { s_val[i] = FLT_MAX; s_idx[i] = 0x7fffffff; }

  float best = valid ? Gb[tid] : FLT_MAX;  // row 0: (0, tid) is upper triangle
  int last = 0;
  if (tid == 0) s_sel[0] = 0;

  for (int it = 1; it < K_; ++it) {
    if (it >= 2) {
      int r = last, c = tid;
      if (r > c) { r = tid; c = last; }
      float s = valid ? Gb[(size_t)r * GP_ + c] : FLT_MAX;
      best = fmaxf(best, s);
    }
    s_val[tid] = valid ? best : FLT_MAX;
    s_idx[tid] = tid;
    __syncthreads();
    for (int off = 256; off >= 1; off >>= 1) {   // argmin, lowest-index tiebreak
      if (tid < off) {
        float v1 = s_val[tid], v2 = s_val[tid + off];
        int   i1 = s_idx[tid], i2 = s_idx[tid + off];
        if (v2 < v1 || (v2 == v1 && i2 < i1)) { s_val[tid] = v2; s_idx[tid] = i2; }
      }
      __syncthreads();
    }
    last = s_idx[0];
    __syncthreads();
    if (tid == 0) s_sel[it] = last;
  }
  __syncthreads();
  if (tid == 0) {                      // sort 64 indices + emit
    for (int i = 1; i < K_; ++i) {
      int key = s_sel[i], j2 = i - 1;
      while (j2 >= 0 && s_sel[j2] > key) { s_sel[j2 + 1] = s_sel[j2]; --j2; }
      s_sel[j2 + 1] = key;
    }
    for (int i = 0; i < K_; ++i) idx[b * K_ + i] = s_sel[i];
  }
}

// ---------------- kernel 4: gather selected rows ----------------
__global__ void __launch_bounds__(256) kgather(const float* __restrict__ video,
                                               const float* __restrict__ audio,
                                               const int* __restrict__ idx,
                                               float* __restrict__ out) {
  int b = blockIdx.y, kk = blockIdx.x, tid = threadIdx.x;
  int row = idx[b * K_ + kk];
  const float* vsrc = video + ((size_t)b * N_ + row) * DV_;
  float* vdst = out + ((size_t)b * K_ + kk) * DV_;
  for (int t = tid; t < DV_; t += 256) vdst[t] = vsrc[t];
  const float* asrc = audio + ((size_t)b * N_ + row) * DA_;
  float* adst = out + (size_t)B_ * K_ * DV_ + ((size_t)b * K_ + kk) * DA_;
  for (int t = tid; t < DA_; t += 256) adst[t] = asrc[t];
}

// ---------------- fallback: self-contained (no workspace, no WMMA) ----------------
__global__ void __launch_bounds__(320) kfallback(const float* __restrict__ video,
                                                 const float* __restrict__ audio,
                                                 float* __restrict__ out) {
  __shared__ float s_val[512];
  __shared__ int   s_idx[512];
  __shared__ int   s_sel[K_];
  __shared__ float s_inv[N_];
  __shared__ float s_row[DV_];
  int b = blockIdx.x, tid = threadIdx.x;
  const float* Vb = video + (size_t)b * N_ * DV_;
  bool valid = tid < N_;

  if (valid) {
    const float* p = Vb + (size_t)tid * DV_;
    float s = 0.f;
    for (int e = 0; e < DV_; ++e) { float x = p[e]; s += x * x; }
    s_inv[tid] = 1.f / (sqrtf(s) + EPS_);
  }
  for (int i = tid + 320; i < 512; i += 320) { s_val[i] = FLT_MAX; s_idx[i] = 0x7fffffff; }
  __syncthreads();

  float myinv = valid ? s_inv[tid] : 0.f;
  float best = FLT_MAX;
  int last = 0;
  if (tid == 0) s_sel[0] = 0;

  for (int it = 1; it < K_; ++it) {
    for (int t = tid; t < DV_; t += 320) s_row[t] = Vb[(size_t)last * DV_ + t];
    __syncthreads();
    if (valid) {
      const float* p = Vb + (size_t)tid * DV_;
      float d = 0.f;
#pragma unroll 4
      for (int e = 0; e < DV_; ++e) d += s_row[e] * p[e];
      float sim = fabsf(d) * s_inv[last] * myinv;
      best = (it == 1) ? sim : fmaxf(best, sim);
    }
    s_val[tid] = valid ? best : FLT_MAX;
    s_idx[tid] = tid;
    __syncthreads();
    for (int off = 256; off >= 1; off >>= 1) {
      if (tid < off) {
        float v1 = s_val[tid], v2 = s_val[tid + off];
        int   i1 = s_idx[tid], i2 = s_idx[tid + off];
        if (v2 < v1 || (v2 == v1 && i2 < i1)) { s_val[tid] = v2; s_idx[tid] = i2; }
      }
      __syncthreads();
    }
    last = s_idx[0];
    __syncthreads();
    if (tid == 0) s_sel[it] = last;
    __syncthreads();
  }
  if (tid == 0) {
    for (int i = 1; i < K_; ++i) {
      int key = s_sel[i], j2 = i - 1;
      while (j2 >= 0 && s_sel[j2] > key) { s_sel[j2 + 1] = s_sel[j2]; --j2; }
      s_sel[j2 + 1] = key;
    }
  }
  __syncthreads();
  for (int kk = 0; kk < K_; ++kk) {
    int row = s_sel[kk];
    const float* vsrc = Vb + (size_t)row * DV_;
    float* vdst = out + ((size_t)b * K_ + kk) * DV_;
    for (int t = tid; t < DV_; t += 320) vdst[t] = vsrc[t];
    const float* asrc = audio + ((size_t)b * N_ + row) * DA_;
    float* adst = out + (size_t)B_ * K_ * DV_ + ((size_t)b * K_ + kk) * DA_;
    for (int t = tid; t < DA_; t += 320) adst[t] = asrc[t];
  }
}

// ---------------- host launcher ----------------
extern "C" void kernel_launch(void* const* d_in, const int* in_sizes, int n_in,
                              void* d_out, int out_size, void* d_ws, size_t ws_size,
                              hipStream_t stream) {
  const float* video = (const float*)d_in[0];
  const float* audio = (const float*)d_in[1];
  float* out = (float*)d_out;

  size_t invBytes = (size_t)B_ * N_ * sizeof(float);        // 153,600
  size_t idxBytes = (size_t)B_ * K_ * sizeof(int);          //  32,768
  size_t gBytes   = (size_t)B_ * GP_ * GP_ * sizeof(float); // ~47.3 MB
  size_t need     = invBytes + idxBytes + gBytes;

  if (ws_size >= need) {
    float* inv = (float*)d_ws;
    int*   idx = (int*)((char*)d_ws + invBytes);
    float* G   = (float*)((char*)d_ws + invBytes + idxBytes);

    knorm  <<<dim3((B_ * N_) / 8), 256, 0, stream>>>(video, inv);
    kgram  <<<dim3((NPAIR_ + 7) / 8, B_), 256, 0, stream>>>(video, inv, G);
    ksel   <<<dim3(B_), 320, 0, stream>>>(G, idx);
    kgather<<<dim3(K_, B_), 256, 0, stream>>>(video, audio, idx, out);
  } else {
    kfallback<<<dim3(B_), 320, 0, stream>>>(video, audio, out);
  }
}